// Decoder_73478300500610
// MI455X (gfx1250) — compile-verified
//
#include <hip/hip_runtime.h>

// out[b,s,h,w] = v[bs] broadcast over the 257x65 grid, where
// v[bs] = sum_d x*W[d] + sin(x)*W[64+d] + cos(x)*W[128+d] + bias.
// Phase 1: 64x192 matvec via V_WMMA_F32_16X16X4_F32 (exact f32 FMA).
// Phase 2: store-bandwidth-bound broadcast fill (4.28 MB ~ 0.18us @ 23.3TB/s).

typedef float v2f __attribute__((ext_vector_type(2)));
typedef float v8f __attribute__((ext_vector_type(8)));

#define GRID_HW 16705   // 257*65
#define NROWS   64      // B*S
#define DDIM    64
#define BPS     8       // blocks per (b,s) slice
#define TPB     256
#define CHUNK   2089    // ceil(16705/8)

__global__ __launch_bounds__(TPB) void decoder_bcast_kernel(
    const float* __restrict__ x,     // [64][64]
    const float* __restrict__ W,     // [192]
    const float* __restrict__ bias,  // [1]
    float* __restrict__ out)         // [64 * 16705]
{
    __shared__ float vsh[NROWS];

    const int tid  = threadIdx.x;
    const int wave = tid >> 5;
    const int lane = tid & 31;

    // ---------- Phase 1: waves 0..3 compute v[0..63] with WMMA ----------
    if (wave < 4) {  // wave-uniform branch: EXEC all 1s inside (WMMA requirement)
        const int half = lane >> 4;        // K-half per ISA A/B f32 layout
        const int m    = lane & 15;        // row within 16-row tile
        const float* xr = x + (wave * 16 + m) * DDIM;

        v8f acc = {0.f, 0.f, 0.f, 0.f, 0.f, 0.f, 0.f, 0.f};

        // identity segment: weights W[0:64]
        for (int kk = 0; kk < 64; kk += 4) {
            const int k0 = kk + 2 * half;
            v2f a = { xr[k0], xr[k0 + 1] };
            v2f b = { W[k0],  W[k0 + 1] };          // all 16 N-columns identical
            acc = __builtin_amdgcn_wmma_f32_16x16x4_f32(false, a, false, b,
                                                        (short)0, acc, false, false);
        }
        // sin segment: weights W[64:128]
        for (int kk = 0; kk < 64; kk += 4) {
            const int k0 = kk + 2 * half;
            v2f a = { __sinf(xr[k0]), __sinf(xr[k0 + 1]) };
            v2f b = { W[64 + k0], W[64 + k0 + 1] };
            acc = __builtin_amdgcn_wmma_f32_16x16x4_f32(false, a, false, b,
                                                        (short)0, acc, false, false);
        }
        // cos segment: weights W[128:192]
        for (int kk = 0; kk < 64; kk += 4) {
            const int k0 = kk + 2 * half;
            v2f a = { __cosf(xr[k0]), __cosf(xr[k0 + 1]) };
            v2f b = { W[128 + k0], W[128 + k0 + 1] };
            acc = __builtin_amdgcn_wmma_f32_16x16x4_f32(false, a, false, b,
                                                        (short)0, acc, false, false);
        }

        // D layout: VGPR j -> M=j (lanes 0-15) / M=j+8 (lanes 16-31); all N equal.
        const float bv = bias[0];
        if (lane == 0) {
            #pragma unroll
            for (int j = 0; j < 8; ++j) vsh[wave * 16 + j] = acc[j] + bv;
        } else if (lane == 16) {
            #pragma unroll
            for (int j = 0; j < 8; ++j) vsh[wave * 16 + 8 + j] = acc[j] + bv;
        }
    }
    __syncthreads();

    // ---------- Phase 2: broadcast fill (store-bound) ----------
    const int s = blockIdx.x / BPS;          // which (b,s) slice
    const int p = blockIdx.x % BPS;          // portion within slice
    const float val = vsh[s];

    const int sliceBase = s * GRID_HW;
    int lo = sliceBase + p * CHUNK;
    int hi = sliceBase + GRID_HW;
    const int hiP = sliceBase + (p + 1) * CHUNK;
    if (hiP < hi) hi = hiP;

    // scalar head to reach 16B alignment (slice bases are odd multiples of 4B)
    int head = (4 - (lo & 3)) & 3;
    if (head > hi - lo) head = hi - lo;
    if (tid < head) out[lo + tid] = val;

    const int b0 = lo + head;
    const int nv = (hi - b0) >> 2;           // aligned float4 stores (B128)
    float4* __restrict__ o4 = (float4*)(out + b0);
    const float4 v4 = make_float4(val, val, val, val);
    for (int i = tid; i < nv; i += TPB) o4[i] = v4;

    // scalar tail
    for (int i = b0 + (nv << 2) + tid; i < hi; i += TPB) out[i] = val;
}

extern "C" void kernel_launch(void* const* d_in, const int* in_sizes, int n_in,
                              void* d_out, int out_size, void* d_ws, size_t ws_size,
                              hipStream_t stream) {
    (void)in_sizes; (void)n_in; (void)d_ws; (void)ws_size; (void)out_size;
    const float* x    = (const float*)d_in[0];  // [8,8,64]
    const float* W    = (const float*)d_in[1];  // [1,192]
    const float* bias = (const float*)d_in[2];  // [1]
    float* out = (float*)d_out;                 // [8,8,257,65]

    decoder_bcast_kernel<<<dim3(NROWS * BPS), dim3(TPB), 0, stream>>>(x, W, bias, out);
}